// GCN_80530636800664
// MI455X (gfx1250) — compile-verified
//
#include <hip/hip_runtime.h>

#define IN_FEAT 128
#define HID 3
#define OUTF 4

typedef __attribute__((ext_vector_type(2)))  float v2f;
typedef __attribute__((ext_vector_type(8)))  float v8f;

// ---------------------------------------------------------------- degree ----
__global__ void k_deg_init(int* __restrict__ cnt, int N) {
  int i = blockIdx.x * blockDim.x + threadIdx.x;
  if (i < N) cnt[i] = 1;  // self-loop contributes 1 to every node's degree
}

__global__ void k_deg_count(const long long* __restrict__ dst, int* __restrict__ cnt, int E) {
  int e = blockIdx.x * blockDim.x + threadIdx.x;
  if (e < E) {
    int d = (int)__builtin_nontemporal_load(&dst[e]);  // streamed once: NT hint
    atomicAdd(&cnt[d], 1);
  }
}

__global__ void k_dinv(const int* __restrict__ cnt, float* __restrict__ dinv, int N) {
  int i = blockIdx.x * blockDim.x + threadIdx.x;
  if (i < N) {
    float d = (float)cnt[i];
    dinv[i] = (d > 0.0f) ? rsqrtf(d) : 0.0f;
  }
}

// ------------------------------------------------- xw = x @ W_gcn (WMMA) ----
// One wave per 16-row tile; K=128 reduced with 32x V_WMMA_F32_16X16X4_F32.
// A 16x4 f32 layout: lane m = lane&15, VGPR0/1 = K pair (2*(lane>>4), +1).
// B 4x16 / C/D 16x16: col = lane&15, row half = lane>>4.
// W (128x3) is staged into LDS zero-padded to 16 columns, pair-interleaved so
// each K-step's B operand is a single aligned ds_load_b64:
//   Wp[(K>>1)*32 + col*2 + (K&1)] = (col<3) ? W[K*3+col] : 0
__global__ void k_xw_wmma(const float* __restrict__ x, const float* __restrict__ Wg,
                          float* __restrict__ xw, int nTiles, int N) {
  __shared__ __align__(16) float Wp[64 * 32];  // 2048 floats = 8 KB

  for (int t = threadIdx.x; t < 64 * 32; t += blockDim.x) {
    int kpair = t >> 5;        // 0..63
    int rem   = t & 31;
    int col   = rem >> 1;      // 0..15
    int K     = kpair * 2 + (rem & 1);
    Wp[t] = (col < HID) ? Wg[K * HID + col] : 0.0f;
  }
  __syncthreads();

  const int lane = threadIdx.x & 31;
  const int wave = threadIdx.x >> 5;
  const int tile = blockIdx.x * (blockDim.x >> 5) + wave;
  if (tile >= nTiles) return;  // wave-uniform; EXEC all-1s at every WMMA

  const int col  = lane & 15;
  const int half = lane >> 4;
  int rowA = tile * 16 + col;
  if (rowA > N - 1) rowA = N - 1;  // branch-free OOB clamp (dup rows never stored)

  const float* __restrict__ xrow = x + (size_t)rowA * IN_FEAT + half * 2;
  const float* __restrict__ wrow = Wp + half * 32 + col * 2;

  v8f acc = {};
#pragma unroll 8
  for (int k0 = 0; k0 < IN_FEAT; k0 += 4) {
    v2f a = *(const v2f*)(xrow + k0);        // global_load_b64, unconditional
    v2f b = *(const v2f*)(wrow + k0 * 16);   // ds_load_b64, unconditional
    acc = __builtin_amdgcn_wmma_f32_16x16x4_f32(false, a, false, b,
                                                (short)0, acc, false, false);
  }

  // Epilogue: lane writes rows (half*8 + v) of its column. The row-range test
  // is wave-uniform (tile is uniform), so full tiles take a branchless path.
  if (col < HID) {
    float* __restrict__ out = xw + (size_t)(tile * 16 + half * 8) * HID + col;
    if (tile * 16 + 16 <= N) {               // uniform: all 16 rows valid
#pragma unroll
      for (int v = 0; v < 8; ++v) out[(size_t)v * HID] = acc[v];
    } else {                                 // tail tile only (absent if N%16==0)
#pragma unroll
      for (int v = 0; v < 8; ++v) {
        int row = tile * 16 + half * 8 + v;
        if (row < N) out[(size_t)v * HID] = acc[v];
      }
    }
  }
}

// ------------------------------------------- agg seed: bias + self-loop ----
__global__ void k_self_init(const float* __restrict__ xw, const float* __restrict__ dinv,
                            const float* __restrict__ bg, float* __restrict__ agg, int N) {
  int i = blockIdx.x * blockDim.x + threadIdx.x;
  if (i >= N) return;
  float di = dinv[i];
  float w = di * di;  // self-loop norm = dinv[i]^2
  agg[i * HID + 0] = bg[0] + xw[i * HID + 0] * w;
  agg[i * HID + 1] = bg[1] + xw[i * HID + 1] * w;
  agg[i * HID + 2] = bg[2] + xw[i * HID + 2] * w;
}

// -------------------------------------------------- edge scatter (atomic) ---
__global__ void k_edge(const long long* __restrict__ src, const long long* __restrict__ dst,
                       const float* __restrict__ dinv, const float* __restrict__ xw,
                       float* __restrict__ agg, int E) {
  int e = blockIdx.x * blockDim.x + threadIdx.x;
  if (e >= E) return;
  int s = (int)__builtin_nontemporal_load(&src[e]);  // 205 MB streamed once; keep
  int d = (int)__builtin_nontemporal_load(&dst[e]);  // L2 for node-indexed arrays
  float w = dinv[s] * dinv[d];
  float m0 = xw[s * HID + 0] * w;
  float m1 = xw[s * HID + 1] * w;
  float m2 = xw[s * HID + 2] * w;
  __hip_atomic_fetch_add(&agg[d * HID + 0], m0, __ATOMIC_RELAXED, __HIP_MEMORY_SCOPE_AGENT);
  __hip_atomic_fetch_add(&agg[d * HID + 1], m1, __ATOMIC_RELAXED, __HIP_MEMORY_SCOPE_AGENT);
  __hip_atomic_fetch_add(&agg[d * HID + 2], m2, __ATOMIC_RELAXED, __HIP_MEMORY_SCOPE_AGENT);
}

// ------------------------------------------- finalize: relu + z = h@W_lin ---
__global__ void k_final(const float* __restrict__ agg, const float* __restrict__ Wl,
                        const float* __restrict__ bl, float* __restrict__ h_out,
                        float* __restrict__ z_out, int N) {
  int i = blockIdx.x * blockDim.x + threadIdx.x;
  if (i >= N) return;
  float h0 = fmaxf(agg[i * HID + 0], 0.0f);
  float h1 = fmaxf(agg[i * HID + 1], 0.0f);
  float h2 = fmaxf(agg[i * HID + 2], 0.0f);
  h_out[i * HID + 0] = h0;
  h_out[i * HID + 1] = h1;
  h_out[i * HID + 2] = h2;
#pragma unroll
  for (int j = 0; j < OUTF; ++j) {
    z_out[i * OUTF + j] = bl[j] + h0 * Wl[0 * OUTF + j]
                                + h1 * Wl[1 * OUTF + j]
                                + h2 * Wl[2 * OUTF + j];
  }
}

// ---------------------------------------------------------------- launch ----
extern "C" void kernel_launch(void* const* d_in, const int* in_sizes, int n_in,
                              void* d_out, int out_size, void* d_ws, size_t ws_size,
                              hipStream_t stream) {
  const float*     x  = (const float*)d_in[0];
  const long long* ei = (const long long*)d_in[1];   // int64 edge_index [2, E]
  const float*     Wg = (const float*)d_in[2];
  const float*     bg = (const float*)d_in[3];
  const float*     Wl = (const float*)d_in[4];
  const float*     bl = (const float*)d_in[5];

  const int N = in_sizes[0] / IN_FEAT;
  const int E = in_sizes[1] / 2;
  const long long* src = ei;
  const long long* dst = ei + E;

  // workspace: cnt[N] int | dinv[N] f32 | xw[3N] f32 | agg[3N] f32  (8N*4 bytes)
  char*  ws   = (char*)d_ws;
  int*   cnt  = (int*)ws;
  float* dinv = (float*)(ws + (size_t)N * 4);
  float* xw   = (float*)(ws + (size_t)N * 8);
  float* agg  = (float*)(ws + (size_t)N * 20);

  float* h_out = (float*)d_out;
  float* z_out = h_out + (size_t)N * HID;

  const int tpb = 256;
  const int nbN = (N + tpb - 1) / tpb;
  const int nbE = (E + tpb - 1) / tpb;

  k_deg_init<<<nbN, tpb, 0, stream>>>(cnt, N);
  k_deg_count<<<nbE, tpb, 0, stream>>>(dst, cnt, E);
  k_dinv<<<nbN, tpb, 0, stream>>>(cnt, dinv, N);

  const int nTiles = (N + 15) / 16;             // 12500 for N=200000 (exact)
  const int wavesPerBlock = 8;                  // 256 threads = 8 wave32
  const int nbT = (nTiles + wavesPerBlock - 1) / wavesPerBlock;
  k_xw_wmma<<<nbT, wavesPerBlock * 32, 0, stream>>>(x, Wg, xw, nTiles, N);

  k_self_init<<<nbN, tpb, 0, stream>>>(xw, dinv, bg, agg, N);
  k_edge<<<nbE, tpb, 0, stream>>>(src, dst, dinv, xw, agg, E);
  k_final<<<nbN, tpb, 0, stream>>>(agg, Wl, bl, h_out, z_out, N);
}